// KNN_1468878815321
// MI455X (gfx1250) — compile-verified
//
#include <hip/hip_runtime.h>
#include <math.h>

typedef float v2f __attribute__((ext_vector_type(2)));
typedef float v8f __attribute__((ext_vector_type(8)));
typedef unsigned int v4u __attribute__((ext_vector_type(4)));
typedef int v8i __attribute__((ext_vector_type(8)));
typedef int v4i __attribute__((ext_vector_type(4)));

#define KNN_K 16
#define NPTS  8192
#define BATCH 4

#define CHUNK_PTS   1024                    // support points per TDM chunk
#define CHUNK_ELEMS (CHUNK_PTS * 3)         // 3072 floats
#define CHUNK_BYTES (CHUNK_ELEMS * 4)       // 12 KB
#define NCHUNK      (NPTS / CHUNK_PTS)      // 8
#define TILES_PER_CHUNK (CHUNK_PTS / 16)    // 64

__device__ __forceinline__ float shflx16_f(float v) { return __shfl_xor(v, 16, 32); }
__device__ __forceinline__ int   shflx16_i(int v)   { return __shfl_xor(v, 16, 32); }

// Issue one TDM copy of CHUNK_BYTES from global (gsrc) into LDS byte offset
// lds_byte_off.  D# built per CDNA5 ISA §8.3/§8.4: 1 descriptor, 4-byte
// elements, single-row tile of CHUNK_ELEMS, type=2 ("image").
__device__ __forceinline__ void tdm_load_chunk(const float* gsrc,
                                               unsigned lds_byte_off) {
  unsigned long long ga = (unsigned long long)(uintptr_t)gsrc;
  v4u g0;
  g0.x = 1u;                                        // count=1, user mode
  g0.y = lds_byte_off;                              // lds_addr (bytes)
  g0.z = (unsigned)(ga & 0xFFFFFFFFu);              // global_addr[31:0]
  g0.w = (unsigned)((ga >> 32) & 0x1FFFFFFu)        // global_addr[56:32]
       | (2u << 30);                                // type=2
  v8i g1;
  g1[0] = (int)(2u << 16);                          // data_size=2 (4B), mask=0
  g1[1] = (int)((CHUNK_ELEMS & 0xFFFFu) << 16);     // tensor_dim0[15:0] @63:48
  g1[2] = (int)(((CHUNK_ELEMS >> 16) & 0xFFFFu)     // tensor_dim0[31:16]
              | (1u << 16));                        // tensor_dim1 = 1
  g1[3] = (int)((CHUNK_ELEMS & 0xFFFFu) << 16);     // tile_dim0 @127:112
  g1[4] = 1;                                        // tile_dim1=1, tile_dim2=0
  g1[5] = (int)CHUNK_ELEMS;                         // tensor_dim0_stride lo32
  g1[6] = 0;                                        // stride hi / dim1_stride lo
  g1[7] = 0;                                        // tensor_dim1_stride hi
  v4i z4 = {0, 0, 0, 0};                            // groups 2/3 unused (<=2D)
  v8i z8 = {0, 0, 0, 0, 0, 0, 0, 0};                // extra group (clang-23 form)
  __builtin_amdgcn_tensor_load_to_lds(g0, g1, z4, z4, z8, 0);
}

__global__ __launch_bounds__(256) void knn_wmma_kernel(
    const float* __restrict__ support,  // [B][N][3]
    const float* __restrict__ query,    // [B][M][3]
    float* __restrict__ out_vals,       // [B][K][M]
    float* __restrict__ out_idx)        // [B][M][K] (indices stored as float)
{
  __shared__ float smem[2 * CHUNK_ELEMS];           // 24 KB double buffer

  const int lane  = threadIdx.x & 31;
  const int wave  = threadIdx.x >> 5;
  const int gwave = blockIdx.x * 8 + wave;   // global wave id
  const int m0g   = gwave * 16;              // global query base
  const int b     = m0g / NPTS;              // uniform within a block
  const int m0    = m0g - b * NPTS;          // query base within batch
  const int col   = lane & 15;
  const bool hi   = lane >= 16;

  // ---- B fragment: query column, norms folded: [-2qx, -2qy, -2qz, 1] ----
  const float* qp = query + ((size_t)b * NPTS + (m0 + col)) * 3;
  const float qx = qp[0], qy = qp[1], qz = qp[2];
  const float q2 = qx * qx + qy * qy + qz * qz;
  v2f bfrag;
  bfrag.x = hi ? (-2.0f * qz) : (-2.0f * qx);
  bfrag.y = hi ? 1.0f         : (-2.0f * qy);

  // ---- running top-K, sorted descending (best[0] = current worst) ----
  float best[KNN_K];
  int   bidx[KNN_K];
#pragma unroll
  for (int j = 0; j < KNN_K; ++j) { best[j] = 3.0e38f; bidx[j] = 0; }

  const float* sb = support + (size_t)b * NPTS * 3;
  const int rowoff = hi ? 8 : 0;
  const unsigned lds_base = (unsigned)(uintptr_t)(void*)smem;  // LDS offset

  // prologue: stage chunk 0 into buffer 0
  if (wave == 0) tdm_load_chunk(sb, lds_base);

  for (int c = 0; c < NCHUNK; ++c) {
    if (wave == 0) {
      if (c + 1 < NCHUNK) {
        tdm_load_chunk(sb + (size_t)(c + 1) * CHUNK_ELEMS,
                       lds_base + (unsigned)((c + 1) & 1) * CHUNK_BYTES);
        __builtin_amdgcn_s_wait_tensorcnt(1);   // chunk c complete
      } else {
        __builtin_amdgcn_s_wait_tensorcnt(0);
      }
    }
    __syncthreads();                            // publish chunk c to all waves

    const float* buf = smem + (c & 1) * CHUNK_ELEMS;
    const int nbase = c * CHUNK_PTS;

    for (int t = 0; t < TILES_PER_CHUNK; ++t) {
      // A fragment from LDS: 16 support rows, K4 = [sx, sy, sz, |s|^2]
      const float* sp = buf + (t * 16 + col) * 3;
      const float sx = sp[0], sy = sp[1], sz = sp[2];
      const float s2 = sx * sx + sy * sy + sz * sz;
      v2f afrag;
      afrag.x = hi ? sz : sx;
      afrag.y = hi ? s2 : sy;

      v8f acc = {};
      acc = __builtin_amdgcn_wmma_f32_16x16x4_f32(
          false, afrag, false, bfrag, (short)0, acc, false, false);

      // acc[v] = |s_n|^2 - 2 <s_n, q_col>  (ranking score)
      const int n00 = nbase + t * 16 + rowoff;
#pragma unroll
      for (int v = 0; v < 8; ++v) {
        const float cnd = acc[v];
        if (cnd < best[0]) {
          best[0] = cnd; bidx[0] = n00 + v;
#pragma unroll
          for (int j = 0; j < KNN_K - 1; ++j) {
            if (best[j] < best[j + 1]) {
              float ts = best[j]; best[j] = best[j + 1]; best[j + 1] = ts;
              int   ti = bidx[j]; bidx[j] = bidx[j + 1]; bidx[j + 1] = ti;
            }
          }
        }
      }
    }
    __syncthreads();                            // all waves done with buffer
  }

  // ---- lane-pair merge (lane, lane^16): exact 16 smallest of 32 ----
  float zs[KNN_K]; int zi[KNN_K];
#pragma unroll
  for (int i = 0; i < KNN_K; ++i) {
    const float ps = shflx16_f(best[i]);
    const int   pi = shflx16_i(bidx[i]);
    const float os = best[KNN_K - 1 - i];
    const int   oi = bidx[KNN_K - 1 - i];
    const bool own = os < ps;
    zs[i] = own ? os : ps;
    zi[i] = own ? oi : pi;
  }
  // bitonic merge of a bitonic 16-sequence -> ascending (nearest first)
#pragma unroll
  for (int d = 8; d >= 1; d >>= 1) {
#pragma unroll
    for (int i = 0; i < KNN_K; ++i) {
      if ((i & d) == 0) {
        const int j = i + d;
        if (zs[i] > zs[j]) {
          float ts = zs[i]; zs[i] = zs[j]; zs[j] = ts;
          int   ti = zi[i]; zi[i] = zi[j]; zi[j] = ti;
        }
      }
    }
  }

  if (!hi) {
    const int m = m0 + col;
    float* vbase = out_vals + (size_t)b * KNN_K * NPTS + m;
    float* ibase = out_idx  + ((size_t)b * NPTS + m) * KNN_K;
#pragma unroll
    for (int k = 0; k < KNN_K; ++k) {
      const float d2 = zs[k] + q2;           // add back query norm
      vbase[(size_t)k * NPTS] = sqrtf(fmaxf(d2, 0.0f));
      ibase[k] = (float)zi[k];
    }
  }
}

extern "C" void kernel_launch(void* const* d_in, const int* in_sizes, int n_in,
                              void* d_out, int out_size, void* d_ws, size_t ws_size,
                              hipStream_t stream) {
  const float* support = (const float*)d_in[0];
  const float* query   = (const float*)d_in[1];
  float* out      = (float*)d_out;
  float* out_vals = out;                                   // B*K*M floats
  float* out_idx  = out + (size_t)BATCH * KNN_K * NPTS;    // B*M*K "ints" as float

  dim3 grid(BATCH * NPTS / (16 * 8));   // 256 blocks
  dim3 block(256);                      // 8 wave32 per block
  hipLaunchKernelGGL(knn_wmma_kernel, grid, block, 0, stream,
                     support, query, out_vals, out_idx);
}